// GptOssTopKRouter_78889959293172
// MI455X (gfx1250) — compile-verified
//
#include <hip/hip_runtime.h>

typedef float v2f __attribute__((ext_vector_type(2)));
typedef float v8f __attribute__((ext_vector_type(8)));

#define HIDDEN      2880
#define NUM_EXPERTS 32
#define TOPK        4
#define WAVES       4
#define K_PER_WAVE  (HIDDEN / WAVES)   // 720
#define STEPS       (K_PER_WAVE / 4)   // 180 WMMA k-steps per wave

// D = A(16x4, tokens x k) * B(4x16, k x experts) + C, fp32 WMMA.
// Per ISA layout: lane holds A[row=lane%16][k0..k0+1] with k0 = kbase + 2*(lane/16),
// and B[k0..k0+1][n=lane%16]; both are contiguous fp32 pairs -> b64 loads.
__global__ __launch_bounds__(128) void gptoss_topk_router_kernel(
    const float* __restrict__ h,      // [T, HIDDEN]
    const float* __restrict__ w,      // [NUM_EXPERTS, HIDDEN]
    const float* __restrict__ bias,   // [NUM_EXPERTS]
    float* __restrict__ out,          // scores [T,32] fp32, then indices [T,4] int32
    int T)
{
    __shared__ float lds[WAVES * 16 * NUM_EXPERTS];   // 8 KB partial logits

    const int tid   = threadIdx.x;
    const int lane  = tid & 31;
    const int wave  = tid >> 5;
    const int row   = lane & 15;      // token-in-tile for A, expert-in-group for B
    const int khalf = lane >> 4;      // which k-pair this lane supplies

    const int t0 = blockIdx.x * 16;
    const int kw = wave * K_PER_WAVE;

    // clamp token for safety (T is a multiple of 16 in this harness)
    const int trow = (t0 + row < T) ? (t0 + row) : (T - 1);

    const float* ha  = h + (size_t)trow       * HIDDEN + kw + 2 * khalf;
    const float* b0p = w + (size_t)row        * HIDDEN + kw + 2 * khalf; // experts 0..15
    const float* b1p = w + (size_t)(16 + row) * HIDDEN + kw + 2 * khalf; // experts 16..31

    v8f acc0 = {0.f, 0.f, 0.f, 0.f, 0.f, 0.f, 0.f, 0.f};
    v8f acc1 = acc0;

#pragma unroll 4
    for (int s = 0; s < STEPS; ++s) {
        const int off = 4 * s;
        v2f a  = *(const v2f*)(ha  + off);
        v2f b0 = *(const v2f*)(b0p + off);
        v2f b1 = *(const v2f*)(b1p + off);
        acc0 = __builtin_amdgcn_wmma_f32_16x16x4_f32(false, a, false, b0,
                                                     (short)0, acc0, false, false);
        acc1 = __builtin_amdgcn_wmma_f32_16x16x4_f32(false, a, false, b1,
                                                     (short)0, acc1, false, false);
    }

    // D layout: VGPR v holds token m = v + 8*(lane/16), expert = group*16 + lane%16
#pragma unroll
    for (int v = 0; v < 8; ++v) {
        const int m = v + 8 * khalf;
        lds[(wave * 16 + m) * NUM_EXPERTS + row]      = acc0[v];
        lds[(wave * 16 + m) * NUM_EXPERTS + 16 + row] = acc1[v];
    }
    __syncthreads();

    // One lane per token: cross-wave reduce + bias, top-4, softmax, scatter.
    if (tid < 16) {
        const int t = t0 + tid;
        if (t < T) {
            float v0 = -3.402823466e38f, v1 = v0, v2 = v0, v3 = v0;
            int   i0 = 0, i1 = 0, i2 = 0, i3 = 0;
#pragma unroll
            for (int e = 0; e < NUM_EXPERTS; ++e) {
                float x = lds[(0 * 16 + tid) * NUM_EXPERTS + e]
                        + lds[(1 * 16 + tid) * NUM_EXPERTS + e]
                        + lds[(2 * 16 + tid) * NUM_EXPERTS + e]
                        + lds[(3 * 16 + tid) * NUM_EXPERTS + e]
                        + bias[e];
                if (x > v0)      { v3=v2;i3=i2; v2=v1;i2=i1; v1=v0;i1=i0; v0=x;i0=e; }
                else if (x > v1) { v3=v2;i3=i2; v2=v1;i2=i1; v1=x;i1=e; }
                else if (x > v2) { v3=v2;i3=i2; v2=x;i2=e; }
                else if (x > v3) { v3=x;i3=e; }
            }
            // softmax over the 4 selected logits (v0 is the max)
            float e1 = __expf(v1 - v0);
            float e2 = __expf(v2 - v0);
            float e3 = __expf(v3 - v0);
            float r  = 1.0f / (1.0f + e1 + e2 + e3);
            const float p0 = r, p1 = e1 * r, p2 = e2 * r, p3 = e3 * r;

            // dense scatter: 32 scores, 4 non-zero
            float4* so = (float4*)(out + (size_t)t * NUM_EXPERTS);
#pragma unroll
            for (int q = 0; q < 8; ++q) {
                float4 val;
                const int eb = q * 4;
                val.x = (eb+0==i0)?p0:(eb+0==i1)?p1:(eb+0==i2)?p2:(eb+0==i3)?p3:0.f;
                val.y = (eb+1==i0)?p0:(eb+1==i1)?p1:(eb+1==i2)?p2:(eb+1==i3)?p3:0.f;
                val.z = (eb+2==i0)?p0:(eb+2==i1)?p1:(eb+2==i2)?p2:(eb+2==i3)?p3:0.f;
                val.w = (eb+3==i0)?p0:(eb+3==i1)?p1:(eb+3==i2)?p2:(eb+3==i3)?p3:0.f;
                so[q] = val;
            }

            int* io = (int*)(out + (size_t)T * NUM_EXPERTS);
            io[(size_t)t * TOPK + 0] = i0;
            io[(size_t)t * TOPK + 1] = i1;
            io[(size_t)t * TOPK + 2] = i2;
            io[(size_t)t * TOPK + 3] = i3;
        }
    }
}

extern "C" void kernel_launch(void* const* d_in, const int* in_sizes, int n_in,
                              void* d_out, int out_size, void* d_ws, size_t ws_size,
                              hipStream_t stream) {
    const float* h    = (const float*)d_in[0];   // [4,4096,2880] fp32
    const float* w    = (const float*)d_in[1];   // [32,2880] fp32
    const float* bias = (const float*)d_in[2];   // [32] fp32
    float* out = (float*)d_out;

    const int T = in_sizes[0] / HIDDEN;          // 16384
    const int blocks = (T + 15) / 16;            // 1024 blocks x 128 threads
    gptoss_topk_router_kernel<<<blocks, 128, 0, stream>>>(h, w, bias, out, T);
}